// BarnesWallQuantizerUnitVol_59785944760426
// MI455X (gfx1250) — compile-verified
//
#include <hip/hip_runtime.h>
#include <hip/hip_bf16.h>

#define TPB 256

// ---------------- compile-time Barnes-Wall tables ----------------
namespace bw {
// Rows of _RM14_G (integer generator of the 32 cosets).
constexpr int GT[5][16] = {
  {1,1,1,1,0,1,0,1,1,0,0,1,0,0,0,0},
  {0,1,1,1,1,0,1,0,1,1,0,0,1,0,0,0},
  {0,0,1,1,1,1,0,1,0,1,1,0,0,1,0,0},
  {0,0,0,1,1,1,1,0,1,0,1,1,0,0,1,0},
  {1,1,1,1,1,1,1,1,1,1,1,1,1,1,1,1}};
// itertools.product bit order: row i selected by bit (4-i) of j.
constexpr int ctab(int j, int k) {
  int v = 0;
  for (int i = 0; i < 5; ++i) v += ((j >> (4 - i)) & 1) * GT[i][k];
  return v;
}
constexpr int mpar(int j)  { int s = 0; for (int k = 0; k < 16; ++k) s += ctab(j,k) >> 1; return s & 1; }
// WHT index of coset j (coeffs of rows 0..3), column permutation k -> 4-bit pattern, inverse.
constexpr int uof(int j) { return ((j>>4)&1) | (((j>>3)&1)<<1) | (((j>>2)&1)<<2) | (((j>>1)&1)<<3); }
constexpr int pat(int k) { int p = 0; for (int i = 0; i < 4; ++i) p |= GT[i][k] << i; return p; }
constexpr int kofp(int p){ for (int k = 0; k < 16; ++k) if (pat(k) == p) return k; return 0; }
// Generator row bitmasks (bit k = coordinate k) and per-coset constant parity mask.
constexpr unsigned gmask(int i){ unsigned m = 0; for (int k = 0; k < 16; ++k) m |= (unsigned)GT[i][k] << k; return m; }
constexpr unsigned mparmask() { unsigned m = 0; for (int j = 0; j < 32; ++j) m |= (unsigned)mpar(j) << j; return m; }
// Flat index into the shared min-butterfly output: [sg][t0][t1][t2][t3].
constexpr int m4idx(int j) {
  return ((j & 1) << 4) | (((j >> 4) & 1) << 3) | (((j >> 3) & 1) << 2)
       | (((j >> 2) & 1) << 1) | ((j >> 1) & 1);
}
} // namespace bw

__global__ __launch_bounds__(TPB) void bw_quant_kernel(
    const float* __restrict__ xin,   // (N,16)
    const float* __restrict__ Crep,  // (32,16)
    const float* __restrict__ a1,    // (1,)
    float* __restrict__ out,         // (N,16)
    int n)
{
    __shared__ float cLds[32 * 16];  // 2 KB codebook (pass-2 gather)
    const int tid = threadIdx.x;

    // ---- CDNA5 async-tensor path: stage codebook global->LDS (128 x 16B).
    if (tid < 128) {
        unsigned ldsOff = (unsigned)(size_t)(void*)cLds + (unsigned)(tid * 16);
        unsigned long long gaddr = (unsigned long long)(Crep + tid * 4);
        asm volatile("global_load_async_to_lds_b128 %0, %1, off"
                     :: "v"(ldsOff), "v"(gaddr) : "memory");
    }
    asm volatile("s_wait_asynccnt 0" ::: "memory");
    __syncthreads();

    const int gid = blockIdx.x * TPB + tid;
    if (gid >= n) return;

    const float av = a1[0];

    // ---- load vector, xh = (x/a)*0.5 (true f32 divide, matches reference)
    float xh[16];
    {
        const float4* p4 = reinterpret_cast<const float4*>(xin + (size_t)gid * 16);
        float4 v0 = p4[0], v1 = p4[1], v2 = p4[2], v3 = p4[3];
        float x16[16] = { v0.x,v0.y,v0.z,v0.w, v1.x,v1.y,v1.z,v1.w,
                          v2.x,v2.y,v2.z,v2.w, v3.x,v3.y,v3.z,v3.w };
#pragma unroll
        for (int k = 0; k < 16; ++k) xh[k] = (x16[k] / av) * 0.5f;
    }

    // ---- precompute the two (even/odd shift) quantizations per coordinate
    float dE[16], dO[16], wq[16];
    float Qe = 0.0f, Fe = 0.0f;
    int   beta = 0;                         // bit k: (f_e - f_o) == 1
#pragma unroll
    for (int k = 0; k < 16; ++k) {
        float te = xh[k];
        float fe = __builtin_rintf(te);
        float re = te - fe;
        float to = xh[k] - 0.5f;
        float fo = __builtin_rintf(to);
        float ro = to - fo;
        float qe = re * re;
        float qo = ro * ro;
        dE[k] = __builtin_fmaf(-2.0f, __builtin_fabsf(re), 1.0f);  // flip cost, even
        dO[k] = __builtin_fmaf(-2.0f, __builtin_fabsf(ro), 1.0f);  // flip cost, odd
        wq[bw::pat(k)] = qo - qe;            // permuted for the WHT
        Qe += qe;
        Fe += fe;
        beta |= ((int)(fe - fo)) << k;       // fe - fo in {0,1}, exact
    }

    // ---- 16-point fast Walsh-Hadamard of (q_o - q_e): all 32 coset D-sums
#pragma unroll
    for (int s = 1; s < 16; s <<= 1) {
#pragma unroll
        for (int i = 0; i < 16; ++i) {
            if (!(i & s)) {
                float A = wq[i], Bv = wq[i | s];
                wq[i]     = A + Bv;
                wq[i | s] = A - Bv;
            }
        }
    }
    const float baseQ = __builtin_fmaf(0.5f, wq[0], Qe);

    // ---- all 32 coset parities packed into one 32-bit word (GF(2)-linear in beta)
    unsigned W = bw::mparmask() ^ (unsigned)(-((int)Fe & 1));
#pragma unroll
    for (int i = 0; i < 5; ++i) {
        constexpr unsigned TM[5] = {0xFFFF0000u, 0xFF00FF00u, 0xF0F0F0F0u,
                                    0xCCCCCCCCu, 0xAAAAAAAAu};
        int hi = __popc(beta & (int)bw::gmask(i)) & 1;
        W ^= (unsigned)(-hi) & TM[i];
    }

    // ---- shared min-butterfly: all 32 coset flip-cost minima in 128 v_min ops
    // Level 1 (pattern bit 3): [q<8][sg][t3]
    float M1[8 * 4];
#pragma unroll
    for (int q = 0; q < 8; ++q)
#pragma unroll
        for (int sg = 0; sg < 2; ++sg)
#pragma unroll
            for (int t3 = 0; t3 < 2; ++t3) {
                float A = sg        ? dO[bw::kofp(q)]     : dE[bw::kofp(q)];
                float B = (sg ^ t3) ? dO[bw::kofp(q + 8)] : dE[bw::kofp(q + 8)];
                M1[q * 4 + sg * 2 + t3] = __builtin_fminf(A, B);
            }
    // Level 2 (bit 2): [q<4][sg][t2][t3]
    float M2[4 * 8];
#pragma unroll
    for (int q = 0; q < 4; ++q)
#pragma unroll
        for (int sg = 0; sg < 2; ++sg)
#pragma unroll
            for (int t2 = 0; t2 < 2; ++t2)
#pragma unroll
                for (int t3 = 0; t3 < 2; ++t3)
                    M2[q * 8 + sg * 4 + t2 * 2 + t3] =
                        __builtin_fminf(M1[q * 4 + sg * 2 + t3],
                                        M1[(q + 4) * 4 + (sg ^ t2) * 2 + t3]);
    // Level 3 (bit 1): [q<2][sg][t1][t2][t3]
    float M3[2 * 16];
#pragma unroll
    for (int q = 0; q < 2; ++q)
#pragma unroll
        for (int sg = 0; sg < 2; ++sg)
#pragma unroll
            for (int t1 = 0; t1 < 2; ++t1)
#pragma unroll
                for (int t2 = 0; t2 < 2; ++t2)
#pragma unroll
                    for (int t3 = 0; t3 < 2; ++t3)
                        M3[q * 16 + sg * 8 + t1 * 4 + t2 * 2 + t3] =
                            __builtin_fminf(M2[q * 8 + sg * 4 + t2 * 2 + t3],
                                            M2[(q + 2) * 8 + (sg ^ t1) * 4 + t2 * 2 + t3]);
    // Level 4 (bit 0): [sg][t0][t1][t2][t3]
    float M4[32];
#pragma unroll
    for (int sg = 0; sg < 2; ++sg)
#pragma unroll
        for (int t0 = 0; t0 < 2; ++t0)
#pragma unroll
            for (int t1 = 0; t1 < 2; ++t1)
#pragma unroll
                for (int t2 = 0; t2 < 2; ++t2)
#pragma unroll
                    for (int t3 = 0; t3 < 2; ++t3)
                        M4[sg * 16 + t0 * 8 + t1 * 4 + t2 * 2 + t3] =
                            __builtin_fminf(M3[0 * 16 + sg * 8 + t1 * 4 + t2 * 2 + t3],
                                            M3[1 * 16 + (sg ^ t0) * 8 + t1 * 4 + t2 * 2 + t3]);

    // ---- pass 1: score all 32 cosets (8 VALU each, everything else shared)
    float bestD = 3.4e38f;
    int   bestJ = 0;
#pragma unroll
    for (int j = 0; j < 32; ++j) {
        const float Dj = (j & 1) ? __builtin_fmaf( 0.5f, wq[bw::uof(j)], baseQ)
                                 : __builtin_fmaf(-0.5f, wq[bw::uof(j)], baseQ);
        const float md  = M4[bw::m4idx(j)];
        const bool  odd = (W & (1u << j)) != 0u;
        const float Dc  = odd ? (Dj + md) : Dj;            // +min(1-2|r|) if odd
        if (Dc < bestD) { bestD = Dc; bestJ = j; }         // strict < == first min
    }

    // ---- pass 2: reference-exact reconstruction of the winning coset
    const float4* c4 = reinterpret_cast<const float4*>(cLds);
    float4 b0 = c4[bestJ * 4 + 0];
    float4 b1 = c4[bestJ * 4 + 1];
    float4 b2 = c4[bestJ * 4 + 2];
    float4 b3 = c4[bestJ * 4 + 3];
    float cb[16] = { b0.x,b0.y,b0.z,b0.w, b1.x,b1.y,b1.z,b1.w,
                     b2.x,b2.y,b2.z,b2.w, b3.x,b3.y,b3.z,b3.w };

    float y[16];
    float p2 = 0.0f, mr2 = -1.0f, rb = 0.0f;
    int   col = 0;
#pragma unroll
    for (int k = 0; k < 16; ++k) {
        float t  = __builtin_fmaf(-0.5f, cb[k], xh[k]);   // 0.5*(x - C), exact
        float f  = __builtin_rintf(t);
        float rv = t - f;
        p2 += f;
        y[k] = __builtin_fmaf(2.0f, f, cb[k]);            // X = 2f + C, exact int
        float m = __builtin_fabsf(rv);
        bool g  = m > mr2;                                // strict > == first max
        mr2 = g ? m  : mr2;
        col = g ? k  : col;
        rb  = g ? rv : rb;
    }
    int ip2 = (int)p2;
    float s = 0.0f;
    if (ip2 & 1) s = (rb > 0.0f) ? 2.0f : ((rb < 0.0f) ? -2.0f : 0.0f);
#pragma unroll
    for (int k = 0; k < 16; ++k) y[k] += (k == col) ? s : 0.0f;

    // ---- store y * a
    float4* o4 = reinterpret_cast<float4*>(out + (size_t)gid * 16);
    o4[0] = make_float4(y[0]  * av, y[1]  * av, y[2]  * av, y[3]  * av);
    o4[1] = make_float4(y[4]  * av, y[5]  * av, y[6]  * av, y[7]  * av);
    o4[2] = make_float4(y[8]  * av, y[9]  * av, y[10] * av, y[11] * av);
    o4[3] = make_float4(y[12] * av, y[13] * av, y[14] * av, y[15] * av);
}

extern "C" void kernel_launch(void* const* d_in, const int* in_sizes, int n_in,
                              void* d_out, int out_size, void* d_ws, size_t ws_size,
                              hipStream_t stream) {
    const float* xin  = (const float*)d_in[0];
    const float* Crep = (const float*)d_in[1];
    const float* a1   = (const float*)d_in[2];
    float* out = (float*)d_out;

    const int n = in_sizes[0] / 16;           // number of 16-d vectors
    const int grid = (n + TPB - 1) / TPB;
    hipLaunchKernelGGL(bw_quant_kernel, dim3(grid), dim3(TPB), 0, stream,
                       xin, Crep, a1, out, n);
}